// DPTreeSeparateOnlyKeyAttention_21079699489131
// MI455X (gfx1250) — compile-verified
//
#include <hip/hip_runtime.h>
#include <hip/hip_bf16.h>
#include <math.h>

typedef __attribute__((ext_vector_type(2))) float v2f;
typedef __attribute__((ext_vector_type(8))) float v8f;

#define EMBED 512
#define HEADS 8
#define HDIM  64
#define SEQ   32
#define TKk   63
#define NSENT 4
#define BSZ   8
#define TQ    64
#define BHn   64   // BSZ*HEADS

__device__ __forceinline__ v8f wmma_f32(v2f a, v2f b, v8f c) {
  // 8 args: (neg_a, A, neg_b, B, c_mod, C, reuse_a, reuse_b)
  return __builtin_amdgcn_wmma_f32_16x16x4_f32(false, a, false, b, (short)0, c, false, false);
}

// ---------------------------------------------------------------------------
// Generic C[M x 512] = (A[M x 512] @ W[512 x 512]^T + bias) * scale
// 64x64 tile per block, 4 waves, each wave: 16 rows x 64 cols.
// ---------------------------------------------------------------------------
__global__ __launch_bounds__(128)
void gemm512_nt(const float* __restrict__ A, const float* __restrict__ W,
                const float* __restrict__ bias, float* __restrict__ C,
                int M, float scale) {
  __shared__ float As[64][33];
  __shared__ float Ws[64][33];
  const int tid  = threadIdx.x;
  const int lane = tid & 31;
  const int wv   = tid >> 5;
  const int bm   = blockIdx.x >> 3;   // 8 column tiles of 64 (N = 512)
  const int bn   = blockIdx.x & 7;
  const int m0   = bm * 64;
  const int n0   = bn * 64;

  v8f acc[4] = {};

  const int rl = (wv << 4) + (lane & 15);
  const int kh = (lane >> 4) << 1;    // 0 or 2

  for (int k0 = 0; k0 < 512; k0 += 32) {
    __syncthreads();
    for (int i = 0; i < 16; i++) {
      int idx = tid + i * 128;        // 0..2047
      int r = idx >> 5, c = idx & 31;
      int gr = m0 + r;
      As[r][c] = (gr < M) ? A[(size_t)gr * 512 + k0 + c] : 0.0f;
      Ws[r][c] = W[(size_t)(n0 + r) * 512 + k0 + c];
    }
    __syncthreads();
    for (int kk = 0; kk < 32; kk += 4) {
      int kb = kk + kh;
      v2f a; a.x = As[rl][kb]; a.y = As[rl][kb + 1];
      for (int nt = 0; nt < 4; nt++) {
        int cl = (nt << 4) + (lane & 15);
        v2f bb; bb.x = Ws[cl][kb]; bb.y = Ws[cl][kb + 1];
        acc[nt] = wmma_f32(a, bb, acc[nt]);
      }
    }
  }

  const int rbase = m0 + (wv << 4) + ((lane >> 4) << 3);
  const int cl = lane & 15;
  for (int nt = 0; nt < 4; nt++) {
    int col = n0 + (nt << 4) + cl;
    float bcol = bias[col];
    for (int vr = 0; vr < 8; vr++) {
      int row = rbase + vr;
      if (row < M) C[(size_t)row * 512 + col] = (acc[nt][vr] + bcol) * scale;
    }
  }
}

// ---------------------------------------------------------------------------
// linear_scores[bh,m,q,kk] = sum_d q[bh,q,d] * k[bh,m,kk,d]   (tk padded->64)
// one block per (bh*4+m); M=64(q), N=64(kk pad), K=64(d)
// ---------------------------------------------------------------------------
__global__ __launch_bounds__(128)
void qk_scores(const float* __restrict__ qb, const float* __restrict__ kb,
               float* __restrict__ sb) {
  __shared__ float Aq[64][33];
  __shared__ float Bk[64][33];
  const int blk = blockIdx.x;
  const int m  = blk & 3;
  const int bh = blk >> 2;
  const int b  = bh >> 3;
  const int h  = bh & 7;
  const int tid  = threadIdx.x;
  const int lane = tid & 31;
  const int wv   = tid >> 5;

  v8f acc[4] = {};
  const int rl = (wv << 4) + (lane & 15);
  const int kh = (lane >> 4) << 1;

  for (int k0 = 0; k0 < 64; k0 += 32) {
    __syncthreads();
    for (int i = 0; i < 16; i++) {
      int idx = tid + i * 128;
      int r = idx >> 5, c = idx & 31;
      Aq[r][c] = qb[(size_t)(r * BSZ + b) * EMBED + h * HDIM + k0 + c];
      Bk[r][c] = (r < TKk)
          ? kb[(size_t)((m * TKk + r) * BSZ + b) * EMBED + h * HDIM + k0 + c]
          : 0.0f;
    }
    __syncthreads();
    for (int kk = 0; kk < 32; kk += 4) {
      int kb2 = kk + kh;
      v2f a; a.x = Aq[rl][kb2]; a.y = Aq[rl][kb2 + 1];
      for (int nt = 0; nt < 4; nt++) {
        int cl = (nt << 4) + (lane & 15);
        v2f bb; bb.x = Bk[cl][kb2]; bb.y = Bk[cl][kb2 + 1];
        acc[nt] = wmma_f32(a, bb, acc[nt]);
      }
    }
  }

  const int rbase = (wv << 4) + ((lane >> 4) << 3);
  const int cl = lane & 15;
  for (int nt = 0; nt < 4; nt++) {
    int col = (nt << 4) + cl;
    for (int vr = 0; vr < 8; vr++) {
      int row = rbase + vr;
      sb[((size_t)blk * 64 + row) * 64 + col] = acc[nt][vr];
    }
  }
}

// ---------------------------------------------------------------------------
// DP tree stage: per (bh,m,q): scatter -> cumsum fw/bw -> 32x32x32 WMMA matmul
// -> gather at span positions -> /sqrt(nonpad), mask -inf.
// Output wb layout: [bh][q][m][64]  (j padded to 64 with -inf)
// ---------------------------------------------------------------------------
__global__ __launch_bounds__(128)
void dp_tree(const float* __restrict__ sb, const int* __restrict__ indices,
             const unsigned char* __restrict__ mask, float* __restrict__ wb) {
  __shared__ int   loS[64], hiS[64];
  __shared__ unsigned char padS[64];
  __shared__ float dinv;
  __shared__ float mat[4][32][33];
  __shared__ float fw[4][32][33];
  __shared__ float bw[4][32][33];

  const int blk = blockIdx.x;   // bh*4+m
  const int m  = blk & 3;
  const int bh = blk >> 2;
  const int b  = bh >> 3;
  const int tid  = threadIdx.x;
  const int lane = tid & 31;
  const int wv   = tid >> 5;

  if (tid < TKk) {
    size_t base = (((size_t)b * NSENT + m) * TKk + tid);
    loS[tid] = indices[base * 2 + 0];
    hiS[tid] = indices[base * 2 + 1];
    padS[tid] = mask[((size_t)b * NSENT + m) * TKk + tid];
  } else if (tid < 64) {
    loS[tid] = 0; hiS[tid] = 0; padS[tid] = 1;
  }
  if (tid == 0) {
    int np = 0;
    for (int j = 0; j < TKk; j++)
      np += mask[((size_t)b * NSENT + m) * TKk + j] ? 0 : 1;
    if (np < 1) np = 1;
    dinv = 1.0f / sqrtf((float)np);
  }
  __syncthreads();

  const int rl = lane & 15;
  const int kh = (lane >> 4) << 1;

  for (int q = wv; q < TQ; q += 4) {
    // zero scatter target
    for (int c = 0; c < 33; c++) mat[wv][lane][c] = 0.0f;
    __syncthreads();
    // scatter the 63 span scores
    for (int jj = lane; jj < TKk; jj += 32) {
      float s = sb[((size_t)blk * 64 + q) * 64 + jj];
      mat[wv][loS[jj]][hiS[jj]] = s;
    }
    __syncthreads();
    // cumsums: fw = row cumsum, bw[i][j] = sum_{r<=j} mat[r][i]
    {
      float run = 0.0f;
      for (int c = 0; c < 32; c++) { run += mat[wv][lane][c]; fw[wv][lane][c] = run; }
      float run2 = 0.0f;
      for (int j = 0; j < 32; j++) { run2 += mat[wv][j][lane]; bw[wv][lane][j] = run2; }
    }
    __syncthreads();
    // dp = fw(32x32) @ bw(32x32) via 4 tiles of WMMA f32 16x16x4
    v8f c00 = {}, c01 = {}, c10 = {}, c11 = {};
    for (int kk = 0; kk < 32; kk += 4) {
      int kb = kk + kh;
      v2f a0; a0.x = fw[wv][rl][kb];      a0.y = fw[wv][rl][kb + 1];
      v2f a1; a1.x = fw[wv][rl + 16][kb]; a1.y = fw[wv][rl + 16][kb + 1];
      v2f b0; b0.x = bw[wv][kb][rl];      b0.y = bw[wv][kb + 1][rl];
      v2f b1; b1.x = bw[wv][kb][rl + 16]; b1.y = bw[wv][kb + 1][rl + 16];
      c00 = wmma_f32(a0, b0, c00);
      c01 = wmma_f32(a0, b1, c01);
      c10 = wmma_f32(a1, b0, c10);
      c11 = wmma_f32(a1, b1, c11);
    }
    __syncthreads();
    // spill dp back to LDS (reuse mat)
    const int rb = (lane >> 4) << 3;
    for (int vr = 0; vr < 8; vr++) {
      mat[wv][rb + vr][rl]           = c00[vr];
      mat[wv][rb + vr][rl + 16]      = c01[vr];
      mat[wv][16 + rb + vr][rl]      = c10[vr];
      mat[wv][16 + rb + vr][rl + 16] = c11[vr];
    }
    __syncthreads();
    // gather + scale + mask
    for (int jj = lane; jj < 64; jj += 32) {
      float outv;
      if (jj < TKk && !padS[jj])
        outv = mat[wv][loS[jj]][hiS[jj]] * dinv;
      else
        outv = -INFINITY;
      wb[(((size_t)bh * TQ + q) * NSENT + m) * 64 + jj] = outv;
    }
    __syncthreads();
  }
}

// ---------------------------------------------------------------------------
// softmax over 256 (padded with -inf -> p=0) then attn = P(64x256) @ V(256x64)
// one block per bh; writes attn in (t, b, e) layout for the final projection.
// ---------------------------------------------------------------------------
__global__ __launch_bounds__(128)
void softmax_attn(const float* __restrict__ wbuf, const float* __restrict__ vb,
                  float* __restrict__ ab) {
  __shared__ float rowMax[64], rowInv[64];
  __shared__ float Pc[64][33];
  __shared__ float Vs[32][65];

  const int bh = blockIdx.x;
  const int b  = bh >> 3;
  const int h  = bh & 7;
  const int tid  = threadIdx.x;
  const int lane = tid & 31;
  const int wv   = tid >> 5;

  // pass 1: per-row max and 1/sum
  if (tid < 64) {
    const float* row = wbuf + ((size_t)bh * TQ + tid) * (NSENT * 64);
    float mx = -INFINITY;
    for (int j = 0; j < NSENT * 64; j++) mx = fmaxf(mx, row[j]);
    float sum = 0.0f;
    for (int j = 0; j < NSENT * 64; j++) sum += __expf(row[j] - mx);
    rowMax[tid] = mx;
    rowInv[tid] = 1.0f / sum;
  }
  __syncthreads();

  v8f acc[4] = {};
  const int rl = (wv << 4) + (lane & 15);
  const int kh = (lane >> 4) << 1;

  for (int c0 = 0; c0 < NSENT * 64; c0 += 32) {
    __syncthreads();
    for (int i = 0; i < 16; i++) {
      int idx = tid + i * 128;            // 2048 total
      { // P chunk: 64 rows x 32 cols
        int r = idx >> 5, c = idx & 31;
        float wvv = wbuf[((size_t)bh * TQ + r) * (NSENT * 64) + c0 + c];
        Pc[r][c] = __expf(wvv - rowMax[r]) * rowInv[r];
      }
      { // V chunk: 32 K-rows x 64 cols
        int r = idx >> 6, c = idx & 63;
        int kg = c0 + r;
        int mm = kg >> 6, j = kg & 63;
        Vs[r][c] = (j < TKk)
            ? vb[(size_t)((mm * TKk + j) * BSZ + b) * EMBED + h * HDIM + c]
            : 0.0f;
      }
    }
    __syncthreads();
    for (int kk = 0; kk < 32; kk += 4) {
      int kb = kk + kh;
      v2f a; a.x = Pc[rl][kb]; a.y = Pc[rl][kb + 1];
      for (int nt = 0; nt < 4; nt++) {
        int cl = (nt << 4) + (lane & 15);
        v2f bb; bb.x = Vs[kb][cl]; bb.y = Vs[kb + 1][cl];
        acc[nt] = wmma_f32(a, bb, acc[nt]);
      }
    }
  }

  const int rbase = (wv << 4) + ((lane >> 4) << 3);
  const int cl = lane & 15;
  for (int nt = 0; nt < 4; nt++) {
    int d = (nt << 4) + cl;
    for (int vr = 0; vr < 8; vr++) {
      int t = rbase + vr;
      ab[((size_t)t * BSZ + b) * EMBED + h * HDIM + d] = acc[nt][vr];
    }
  }
}

// ---------------------------------------------------------------------------
extern "C" void kernel_launch(void* const* d_in, const int* in_sizes, int n_in,
                              void* d_out, int out_size, void* d_ws, size_t ws_size,
                              hipStream_t stream) {
  (void)in_sizes; (void)n_in; (void)out_size; (void)ws_size;
  const float* query = (const float*)d_in[0];          // (64, 8, 512)
  const float* key   = (const float*)d_in[1];          // (4, 63, 8, 512)
  const int*   idx   = (const int*)d_in[2];            // (8, 4, 63, 2)
  const unsigned char* mask = (const unsigned char*)d_in[3]; // (8, 4, 63) bool
  const float* ipw   = (const float*)d_in[4];          // (1536, 512)
  const float* ipb   = (const float*)d_in[5];          // (1536,)
  const float* ow    = (const float*)d_in[6];          // (512, 512)
  const float* ob    = (const float*)d_in[7];          // (512,)
  float* out = (float*)d_out;

  float* ws = (float*)d_ws;
  float* qb = ws;                        // 512*512
  float* kb = qb + 512 * 512;            // 2016*512
  float* vb = kb + 2016 * 512;           // 2016*512
  float* sb = vb + 2016 * 512;           // 256*64*64
  float* wb = sb + 256 * 64 * 64;        // 64*64*4*64
  float* ab = wb + 64 * 64 * 4 * 64;     // 512*512

  dim3 blk(128);
  const float scaling = 0.125f;          // HEAD_DIM^-0.5

  gemm512_nt<<<64,  blk, 0, stream>>>(query, ipw,                 ipb,        qb, 512,  scaling);
  gemm512_nt<<<256, blk, 0, stream>>>(key,   ipw + 512 * 512,     ipb + 512,  kb, 2016, 1.0f);
  gemm512_nt<<<256, blk, 0, stream>>>(key,   ipw + 2 * 512 * 512, ipb + 1024, vb, 2016, 1.0f);
  qk_scores  <<<256, blk, 0, stream>>>(qb, kb, sb);
  dp_tree    <<<256, blk, 0, stream>>>(sb, idx, mask, wb);
  softmax_attn<<<64, blk, 0, stream>>>(wb, vb, ab);
  gemm512_nt<<<64,  blk, 0, stream>>>(ab, ow, ob, out, 512, 1.0f);
}